// SpatialVolumeNet_12532714570268
// MI455X (gfx1250) — compile-verified
//
#include <hip/hip_runtime.h>

// ---------------------------------------------------------------------------
// SpatialVolumeNet for MI455X (gfx1250, wave32)
//   setup kernel : 1 wave; P = K3x3 @ pose via V_WMMA_F32_16X16X4_F32 (16 tiny
//                  GEMMs on the matrix pipe), then 3x3 inverse + camera dist.
//   main kernel  : one wave per (bt,d,h) output row; trilinear gather from the
//                  L2-resident volume with paired-x b64 gathers, NT stores.
// ---------------------------------------------------------------------------

typedef float v2f __attribute__((ext_vector_type(2)));
typedef float v8f __attribute__((ext_vector_type(8)));

#define VDIM   32
#define CCH    64
#define DD     48
#define HH     32
#define WW     32
#define TNV    8
#define NNV    16
#define BTV    16
#define FVL_C  0.86603f

// feats elements = 16*64*48*32*32
#define FEATS_ELEMS 50331648ull
#define VOL_CH      32768        // 32^3 elements per channel
#define SPW         49152        // D*H*W = channel stride in feats

// 4-byte-aligned float pair (x-adjacent corner values)
struct __attribute__((packed, aligned(4))) f2u { float x, y; };

// ---------------------------------------------------------------------------
// Setup: per (b,tn) view, compute Minv (3x3), translation, near/far -> d_ws
// ---------------------------------------------------------------------------
__global__ __launch_bounds__(32)
void svn_setup(const float* __restrict__ RT,   // (B,N,3,4)
               const float* __restrict__ Km,   // (B,N,4,4)
               const int*   __restrict__ tidx, // (B,TN)
               float*       __restrict__ params)
{
    __shared__ float Pbuf[BTV][12];
    const int L  = threadIdx.x;          // 0..31, single wave
    const int m  = L & 15;
    const int k0 = (L < 16) ? 0 : 2;

    // Phase 1: P = K3x3 @ pose(3x4), one 16x16x4 WMMA per view (uniform loop,
    // EXEC all ones as required by WMMA).
    for (int bt = 0; bt < BTV; ++bt) {
        const int b  = bt >> 3;
        const int tn = bt & 7;
        const int n  = tidx[b * TNV + tn];
        const float* K3 = Km + (size_t)(b * NNV + n) * 16;  // 4x4 row-major
        const float* ps = RT + (size_t)(b * NNV + n) * 12;  // 3x4 row-major

        // A (16x4 f32): rows = K rows (3 valid), K dim = K cols (3 valid)
        v2f a;
        a.x = (m < 3 && k0     < 3) ? K3[m * 4 + k0]     : 0.0f;
        a.y = (m < 3 && (k0+1) < 3) ? K3[m * 4 + k0 + 1] : 0.0f;
        // B (4x16 f32): K dim = pose rows (3 valid), cols = pose cols (4 valid)
        v2f bb;
        bb.x = (k0     < 3 && m < 4) ? ps[k0 * 4 + m]       : 0.0f;
        bb.y = ((k0+1) < 3 && m < 4) ? ps[(k0 + 1) * 4 + m] : 0.0f;

#if __has_builtin(__builtin_amdgcn_wmma_f32_16x16x4_f32)
        v8f cz = {};
        v8f dd = __builtin_amdgcn_wmma_f32_16x16x4_f32(
            false, a, false, bb, (short)0, cz, false, false);
        // D layout: lane L<16 holds column N=L, rows M=0..7 in dd[0..7]
        if (L < 4) {
            Pbuf[bt][0 * 4 + L] = dd[0];
            Pbuf[bt][1 * 4 + L] = dd[1];
            Pbuf[bt][2 * 4 + L] = dd[2];
        }
#else
        if (L < 12) {
            const int i = L >> 2, j = L & 3;
            float s = 0.0f;
            for (int k = 0; k < 3; ++k) s += K3[i * 4 + k] * ps[k * 4 + j];
            Pbuf[bt][i * 4 + j] = s;
        }
#endif
    }
    __syncthreads();

    // Phase 2: lanes 0..15, one view each: invert M, translation, near/far.
    if (L < BTV) {
        const int bt = L;
        const int b  = bt >> 3;
        const int tn = bt & 7;
        const int n  = tidx[b * TNV + tn];
        const float* P = Pbuf[bt];
        const float M00 = P[0], M01 = P[1],  M02 = P[2],  t0 = P[3];
        const float M10 = P[4], M11 = P[5],  M12 = P[6],  t1 = P[7];
        const float M20 = P[8], M21 = P[9],  M22 = P[10], t2 = P[11];

        const float c00 = M11*M22 - M12*M21;
        const float c01 = M02*M21 - M01*M22;
        const float c02 = M01*M12 - M02*M11;
        const float c10 = M12*M20 - M10*M22;
        const float c11 = M00*M22 - M02*M20;
        const float c12 = M02*M10 - M00*M12;
        const float c20 = M10*M21 - M11*M20;
        const float c21 = M01*M20 - M00*M21;
        const float c22 = M00*M11 - M01*M10;
        const float id  = 1.0f / (M00*c00 + M01*c10 + M02*c20);

        const float i00 = c00*id, i01 = c01*id, i02 = c02*id;
        const float i10 = c10*id, i11 = c11*id, i12 = c12*id;
        const float i20 = c20*id, i21 = c21*id, i22 = c22*id;

        const float r0 = -(i00*t0 + i01*t1 + i02*t2);
        const float r1 = -(i10*t0 + i11*t1 + i12*t2);
        const float r2 = -(i20*t0 + i21*t1 + i22*t2);

        // camera distance from the selected pose: cam_i = -sum_j R[j][i]*T[j]
        const float* ps = RT + (size_t)(b * NNV + n) * 12;
        const float cp0 = -(ps[0]*ps[3] + ps[4]*ps[7] + ps[8]*ps[11]);
        const float cp1 = -(ps[1]*ps[3] + ps[5]*ps[7] + ps[9]*ps[11]);
        const float cp2 = -(ps[2]*ps[3] + ps[6]*ps[7] + ps[10]*ps[11]);
        const float dist = sqrtf(cp0*cp0 + cp1*cp1 + cp2*cp2);

        float* o = params + bt * 16;
        o[0] = i00; o[1] = i01; o[2]  = i02;
        o[3] = i10; o[4] = i11; o[5]  = i12;
        o[6] = i20; o[7] = i21; o[8]  = i22;
        o[9] = r0;  o[10] = r1; o[11] = r2;
        o[12] = dist - FVL_C;   // near
        o[13] = dist + FVL_C;   // far
        o[14] = 0.0f; o[15] = 0.0f;
    }
}

// ---------------------------------------------------------------------------
// Main: one lane per output pixel; wave = one (bt,d,h) row of 32 pixels.
// ---------------------------------------------------------------------------
__global__ __launch_bounds__(256)
void svn_main(const float* __restrict__ vol,    // (B,C,32,32,32)
              const float* __restrict__ params, // (BT,16)
              float*       __restrict__ out)
{
    const int p  = blockIdx.x * 256 + threadIdx.x;
    const int w  = p & 31;
    const int h  = (p >> 5) & 31;
    const int dh = p >> 10;                 // = bt*48 + d, wave-uniform
    int d  = dh % DD;
    int bt = dh / DD;
    bt = __builtin_amdgcn_readfirstlane(bt);
    d  = __builtin_amdgcn_readfirstlane(d);

    const float* pr = params + bt * 16;     // scalar loads (uniform)
    const float i00 = pr[0], i01 = pr[1], i02 = pr[2];
    const float i10 = pr[3], i11 = pr[4], i12 = pr[5];
    const float i20 = pr[6], i21 = pr[7], i22 = pr[8];
    const float r0  = pr[9], r1  = pr[10], r2 = pr[11];
    const float nearv = pr[12], farv = pr[13];

    // ray point for this pixel at this depth slice
    const float t     = (float)d * (1.0f / 47.0f);
    const float depth = nearv + t * (farv - nearv);
    const float px = (float)(w * 8) * depth;   // IMG/FS = 8
    const float py = (float)(h * 8) * depth;
    const float pz = depth;

    // grid coords: (Minv @ pts + tr) / SVL, SVL = 0.5
    const float gx = (i00*px + i01*py + i02*pz + r0) * 2.0f;
    const float gy = (i10*px + i11*py + i12*pz + r1) * 2.0f;
    const float gz = (i20*px + i21*py + i22*pz + r2) * 2.0f;

    // align_corners=True: x = (g+1)*0.5*(V-1)
    const float X = (gx + 1.0f) * 15.5f;
    const float Y = (gy + 1.0f) * 15.5f;
    const float Z = (gz + 1.0f) * 15.5f;

    const float x0f = floorf(X), y0f = floorf(Y), z0f = floorf(Z);
    const float xd = X - x0f, yd = Y - y0f, zd = Z - z0f;

    // per-axis weights with zero-padding mask folded in (float-domain checks:
    // safe for huge/NaN coords -> weights collapse to 0 like the reference)
    const float wx0 = (x0f >=  0.0f && x0f <= 31.0f) ? (1.0f - xd) : 0.0f;
    const float wx1 = (x0f >= -1.0f && x0f <= 30.0f) ? xd          : 0.0f;
    const float wy0 = (y0f >=  0.0f && y0f <= 31.0f) ? (1.0f - yd) : 0.0f;
    const float wy1 = (y0f >= -1.0f && y0f <= 30.0f) ? yd          : 0.0f;
    const float wz0 = (z0f >=  0.0f && z0f <= 31.0f) ? (1.0f - zd) : 0.0f;
    const float wz1 = (z0f >= -1.0f && z0f <= 30.0f) ? zd          : 0.0f;

    // clamped indices; x corners fetched as one 8B pair at xb in [0,30]
    const int xb  = (int)fminf(fmaxf(x0f,        0.0f), 30.0f);
    const int yc0 = (int)fminf(fmaxf(y0f,        0.0f), 31.0f);
    const int yc1 = (int)fminf(fmaxf(y0f + 1.0f, 0.0f), 31.0f);
    const int zc0 = (int)fminf(fmaxf(z0f,        0.0f), 31.0f);
    const int zc1 = (int)fminf(fmaxf(z0f + 1.0f, 0.0f), 31.0f);
    const bool selHi = (x0f > 30.0f);   // x0 corner sits in pair.hi
    const bool selLo = (x0f <  0.0f);   // x1 corner sits in pair.lo

    const int off0 = (zc0 * 32 + yc0) * 32 + xb;
    const int off1 = (zc0 * 32 + yc1) * 32 + xb;
    const int off2 = (zc1 * 32 + yc0) * 32 + xb;
    const int off3 = (zc1 * 32 + yc1) * 32 + xb;

    const float wA0 = wz0*wy0*wx0, wB0 = wz0*wy0*wx1;
    const float wA1 = wz0*wy1*wx0, wB1 = wz0*wy1*wx1;
    const float wA2 = wz1*wy0*wx0, wB2 = wz1*wy0*wx1;
    const float wA3 = wz1*wy1*wx0, wB3 = wz1*wy1*wx1;

    const float* vb = vol + (size_t)(bt >> 3) * (CCH * VOL_CH);
    float* fout = out + (size_t)bt * (CCH * (size_t)SPW)
                      + (size_t)(d * 1024 + h * 32 + w);

    #pragma unroll 4
    for (int c = 0; c < CCH; ++c) {
        const float* vc = vb + (size_t)c * VOL_CH;
        const f2u p0 = *(const f2u*)(vc + off0);
        const f2u p1 = *(const f2u*)(vc + off1);
        const f2u p2 = *(const f2u*)(vc + off2);
        const f2u p3 = *(const f2u*)(vc + off3);
        const float a0 = selHi ? p0.y : p0.x,  b0 = selLo ? p0.x : p0.y;
        const float a1 = selHi ? p1.y : p1.x,  b1 = selLo ? p1.x : p1.y;
        const float a2 = selHi ? p2.y : p2.x,  b2 = selLo ? p2.x : p2.y;
        const float a3 = selHi ? p3.y : p3.x,  b3 = selLo ? p3.x : p3.y;

        float acc = wA0 * a0;
        acc = fmaf(wB0, b0, acc);
        acc = fmaf(wA1, a1, acc);
        acc = fmaf(wB1, b1, acc);
        acc = fmaf(wA2, a2, acc);
        acc = fmaf(wB2, b2, acc);
        acc = fmaf(wA3, a3, acc);
        acc = fmaf(wB3, b3, acc);

        // stream output past L2 so the 16.8MB volume stays resident
        __builtin_nontemporal_store(acc, fout + (size_t)c * SPW);
    }

    // depth_vol output (second tuple element), value depends on (bt,d) only
    float* dout = out + FEATS_ELEMS + (size_t)bt * SPW
                      + (size_t)(d * 1024 + h * 32 + w);
    __builtin_nontemporal_store(depth, dout);
}

// ---------------------------------------------------------------------------
extern "C" void kernel_launch(void* const* d_in, const int* in_sizes, int n_in,
                              void* d_out, int out_size, void* d_ws, size_t ws_size,
                              hipStream_t stream)
{
    const float* vol = (const float*)d_in[0];  // spatial_volume (2,64,32,32,32)
    const float* RT  = (const float*)d_in[1];  // target_RT (2,16,3,4)
    const float* Km  = (const float*)d_in[2];  // target_K  (2,16,4,4)
    const int*   ti  = (const int*)  d_in[3];  // target_indices (2,8)
    float* out    = (float*)d_out;
    float* params = (float*)d_ws;              // 16 views * 16 floats

    svn_setup<<<1, 32, 0, stream>>>(RT, Km, ti, params);

    // 16*48*32*32 / 256 = 3072 blocks, 8 waves each; wave = one output row
    svn_main<<<3072, 256, 0, stream>>>(vol, params, out);
}